// TripletLossBatchAll_82489141887188
// MI455X (gfx1250) — compile-verified
//
#include <hip/hip_runtime.h>
#include <hip/hip_bf16.h>

// ---------------------------------------------------------------------------
// TripletLossBatchAll for n=512, d=512 on MI455X (gfx1250, wave32).
//   1) sq[i]      = ||X_i||^2                          (rowsq kernel)
//   2) dist[i][j] = sqrt(clip(sq_i + sq_j - 2*X X^T))  (WMMA f32 16x16x4 GEMM)
//   3) loss       = mean relu(d_ap - d_an + 0.3) over valid (a,p,n)
//      per-anchor LDS compaction of the positive set -> ~|P|*|N| work only.
// ---------------------------------------------------------------------------

typedef __attribute__((ext_vector_type(2))) float v2f;
typedef __attribute__((ext_vector_type(8))) float v8f;

#define N_SAMP 512
#define MARGIN 0.3f
#define EPS_D  1e-12f

// -------------------------------- init ------------------------------------
__global__ void tl_init_accum(float* accum) {
    if (threadIdx.x == 0) { accum[0] = 0.0f; accum[1] = 0.0f; }
}

// ----------------------------- row norms ----------------------------------
__global__ __launch_bounds__(256) void tl_rowsq(const float* __restrict__ X,
                                                float* __restrict__ sq) {
    __shared__ float red[256];
    const int row = blockIdx.x;
    float s = 0.0f;
    for (int j = threadIdx.x; j < N_SAMP; j += 256) {
        float v = X[row * N_SAMP + j];
        s += v * v;
    }
    red[threadIdx.x] = s;
    __syncthreads();
    for (int off = 128; off > 0; off >>= 1) {
        if (threadIdx.x < off) red[threadIdx.x] += red[threadIdx.x + off];
        __syncthreads();
    }
    if (threadIdx.x == 0) sq[row] = red[0];
}

// --------------------- WMMA fp32 GEMM -> distance matrix -------------------
// One wave32 computes one 16x16 tile of G = X*X^T via V_WMMA_F32_16X16X4_F32.
// A-matrix (16x4): lanes 0-15 -> M=lane, K={k,k+1}; lanes 16-31 -> K={k+2,k+3}.
// B-matrix (4x16): same per-lane addressing with the tile-column rows of X
// (B[k][n] = X[col_n][k], i.e. X^T column-major == X row-major).
// C/D 16x16 f32: VGPR r, lane L -> M = r + 8*(L>>4), N = L&15.
__global__ __launch_bounds__(128) void tl_gemm_dist(const float* __restrict__ X,
                                                    const float* __restrict__ sq,
                                                    float* __restrict__ dist) {
    const int wave = threadIdx.x >> 5;             // 0..3
    const int lane = threadIdx.x & 31;
    const int tile = blockIdx.x * 4 + wave;        // 0..1023
    const int tm   = tile >> 5;                    // tile row (0..31)
    const int tn   = tile & 31;                    // tile col (0..31)

    const int half = lane >> 4;                    // 0: K pair {0,1}, 1: {2,3}
    const int l16  = lane & 15;

    const float* __restrict__ arow = X + (size_t)(tm * 16 + l16) * N_SAMP + half * 2;
    const float* __restrict__ brow = X + (size_t)(tn * 16 + l16) * N_SAMP + half * 2;

    v8f c = {};
#pragma unroll 16
    for (int k = 0; k < N_SAMP; k += 4) {
        v2f av = *(const v2f*)(arow + k);          // 8B aligned
        v2f bv = *(const v2f*)(brow + k);
        // 8 args: (neg_a, A, neg_b, B, c_mod, C, reuse_a, reuse_b)
        c = __builtin_amdgcn_wmma_f32_16x16x4_f32(
                false, av, false, bv, (short)0, c, false, false);
    }

    const int nGlob = tn * 16 + l16;
    const float sqn = sq[nGlob];
#pragma unroll
    for (int r = 0; r < 8; ++r) {
        const int mGlob = tm * 16 + r + 8 * half;
        float d2 = sq[mGlob] + sqn - 2.0f * c[r];
        d2 = d2 < EPS_D ? EPS_D : d2;
        // native v_sqrt_f32: ~1 ulp, plenty for a +/-0.3 margin comparison
        dist[(size_t)mGlob * N_SAMP + nGlob] = __builtin_amdgcn_sqrtf(d2);
    }
}

// --------------------- per-anchor triplet reduction ------------------------
// Block = anchor. Compact positive indices into LDS (|P|~8), then each thread
// strides over negatives: sum += relu(d_ap - d_an + margin) for all p in P.
__global__ __launch_bounds__(256) void tl_reduce(const float* __restrict__ dist,
                                                 const int* __restrict__ targets,
                                                 float* __restrict__ accum) {
    __shared__ float drow[N_SAMP];
    __shared__ int   tgt[N_SAMP];
    __shared__ int   plist[N_SAMP];
    __shared__ int   npos_s;
    __shared__ float red[256];

    const int a = blockIdx.x;
    if (threadIdx.x == 0) npos_s = 0;
    for (int i = threadIdx.x; i < N_SAMP; i += 256) {
        drow[i] = dist[(size_t)a * N_SAMP + i];
        tgt[i]  = targets[i];
    }
    __syncthreads();

    const int ta = tgt[a];
    for (int i = threadIdx.x; i < N_SAMP; i += 256) {
        if (tgt[i] == ta) {                        // diagonal included, like ref
            int idx = atomicAdd(&npos_s, 1);
            plist[idx] = i;
        }
    }
    __syncthreads();
    const int npos = npos_s;

    float sum = 0.0f;
    int nneg_local = 0;
    for (int n = threadIdx.x; n < N_SAMP; n += 256) {
        if (tgt[n] != ta) {
            ++nneg_local;
            const float dan = drow[n];
            for (int pi = 0; pi < npos; ++pi) {
                float v = drow[plist[pi]] - dan + MARGIN;
                sum += v > 0.0f ? v : 0.0f;
            }
        }
    }

    // block reduce sum
    red[threadIdx.x] = sum;
    __syncthreads();
    for (int off = 128; off > 0; off >>= 1) {
        if (threadIdx.x < off) red[threadIdx.x] += red[threadIdx.x + off];
        __syncthreads();
    }
    const float block_sum = red[0];
    __syncthreads();

    // block reduce negative count
    red[threadIdx.x] = (float)nneg_local;
    __syncthreads();
    for (int off = 128; off > 0; off >>= 1) {
        if (threadIdx.x < off) red[threadIdx.x] += red[threadIdx.x + off];
        __syncthreads();
    }

    if (threadIdx.x == 0) {
        atomicAdd(&accum[0], block_sum);
        atomicAdd(&accum[1], red[0] * (float)npos);  // |P_a| * |N_a|
    }
}

// ------------------------------- finalize ----------------------------------
__global__ void tl_finalize(const float* accum, float* out) {
    if (threadIdx.x == 0) out[0] = accum[0] / accum[1];
}

// ---------------------------------------------------------------------------
extern "C" void kernel_launch(void* const* d_in, const int* in_sizes, int n_in,
                              void* d_out, int out_size, void* d_ws, size_t ws_size,
                              hipStream_t stream) {
    const float* X       = (const float*)d_in[0];   // [512,512] fp32
    const int*   targets = (const int*)d_in[1];     // [512] int
    float*       out     = (float*)d_out;           // scalar loss

    float* dist  = (float*)d_ws;                    // 512*512 floats (1 MB)
    float* sq    = dist + (size_t)N_SAMP * N_SAMP;  // 512 floats
    float* accum = sq + N_SAMP;                     // {sum, count}

    tl_init_accum<<<1, 32, 0, stream>>>(accum);
    tl_rowsq<<<N_SAMP, 256, 0, stream>>>(X, sq);
    tl_gemm_dist<<<256, 128, 0, stream>>>(X, sq, dist);   // 1024 waves = 1024 tiles
    tl_reduce<<<N_SAMP, 256, 0, stream>>>(dist, targets, accum);
    tl_finalize<<<1, 32, 0, stream>>>(accum, out);
}